// Conv2dLocal_47132971106931
// MI455X (gfx1250) — compile-verified
//
#include <hip/hip_runtime.h>
#include <hip/hip_bf16.h>

// Problem constants (from reference)
#define CH     128
#define HW     256
#define LTOT   65536                 // 256*256 output positions
#define KK     9                     // 3x3 kernel
#define TILE   256                   // l' values per block
#define WBYTES (TILE * KK * 4)       // 9216 B of weights per block
#define NCHUNK (WBYTES / 16)         // 576 x 16B async chunks
#define NLINES (WBYTES / 128)        // 72 x 128B prefetch lines

// 12-byte row fragment; align 4 so the backend may merge to b96/b64+b32.
struct alignas(4) F3 { float a, b, c; };

__global__ __launch_bounds__(256)
void conv2d_local_gfx1250(const float* __restrict__ x,
                          const float* __restrict__ w,
                          float* __restrict__ out) {
    __shared__ __align__(16) float wlds[TILE * KK];   // 9216 B

    const int tid = threadIdx.x;          // 0..255
    const int cp  = blockIdx.y;           // c' : 0..127
    const int B0  = blockIdx.x * TILE;    // l' tile base

    // ---------------- Async stage: 9216 contiguous weight bytes -> LDS ----
    // W[B0 + t][k] lands at wlds[t*9 + k]; slab base is 16B aligned
    // (B0 multiple of 256 -> B0*36 bytes multiple of 16).
    const float* wbase = w + (size_t)B0 * KK;              // uniform -> SGPR64
    const unsigned lds0 = (unsigned)(unsigned long long)(&wlds[0]);
    #pragma unroll
    for (int r = 0; r < 3; ++r) {
        const int chunk = tid + r * 256;
        if (chunk < NCHUNK) {
            const unsigned goff  = (unsigned)chunk * 16u;  // byte offset in slab
            const unsigned laddr = lds0 + goff;            // LDS byte address
            asm volatile("global_load_async_to_lds_b128 %0, %1, %2"
                         :: "v"(laddr), "v"(goff), "s"(wbase)
                         : "memory");
        }
    }

    // Speculative prefetch of next tile's weight slab (global_prefetch_b8).
    if ((blockIdx.x + 1u) < gridDim.x && tid < NLINES) {
        __builtin_prefetch((const char*)wbase + WBYTES + tid * 128, 0, 0);
    }

    // Drain this wave's async copies; barrier makes all waves' LDS visible.
    asm volatile("s_wait_asynccnt 0" ::: "memory");
    __syncthreads();

    // ---------------- Index scramble from the raw reshape ----------------
    const int lp = B0 + tid;                    // l'
    const int c  = lp & (CH - 1);               // input channel
    const int l  = (cp << 9) + (lp >> 7);       // scrambled spatial index
    const int oh = l >> 8;                      // wave-uniform
    const int ow = l & (HW - 1);                // wave-uniform

    // Per-thread 9 weights from LDS (stride 9 across lanes: 9 coprime with
    // 64 banks -> conflict-free).
    float wk[KK];
    #pragma unroll
    for (int k = 0; k < KK; ++k) wk[k] = wlds[tid * KK + k];

    const float* xc = x + (size_t)c * (HW * HW);
    float acc = 0.0f;

    // (oh, ow) are wave-uniform, so this is a scalar (non-divergent) branch.
    if (oh >= 1 && oh <= HW - 2 && ow >= 1 && ow <= HW - 2) {
        // Interior fast path: 3 unconditional 12B row loads per thread.
        const float* p = xc + (oh - 1) * HW + (ow - 1);
        #pragma unroll
        for (int di = 0; di < 3; ++di) {
            const F3 r = *(const F3*)(p + di * HW);
            acc = fmaf(r.a, wk[di * 3 + 0], acc);
            acc = fmaf(r.b, wk[di * 3 + 1], acc);
            acc = fmaf(r.c, wk[di * 3 + 2], acc);
        }
    } else {
        // Boundary path: zero padding via unsigned-compare bounds checks.
        #pragma unroll
        for (int di = 0; di < 3; ++di) {
            const int h = oh + di - 1;
            const bool hin = (unsigned)h < (unsigned)HW;
            const float* row = xc + h * HW;
            #pragma unroll
            for (int dj = 0; dj < 3; ++dj) {
                const int wc = ow + dj - 1;
                const float v =
                    (hin && (unsigned)wc < (unsigned)HW) ? row[wc] : 0.0f;
                acc = fmaf(v, wk[di * 3 + dj], acc);
            }
        }
    }

    // Lane-coalesced, non-temporal store: output is write-once, keep it from
    // evicting the L2-resident input/weight working set.
    __builtin_nontemporal_store(acc, &out[(size_t)cp * LTOT + lp]);
}

extern "C" void kernel_launch(void* const* d_in, const int* in_sizes, int n_in,
                              void* d_out, int out_size, void* d_ws, size_t ws_size,
                              hipStream_t stream) {
    (void)in_sizes; (void)n_in; (void)d_ws; (void)ws_size; (void)out_size;
    const float* x = (const float*)d_in[0];   // (1,128,256,256) fp32
    const float* w = (const float*)d_in[1];   // (65536,9) fp32
    float* o = (float*)d_out;                 // (1,128,256,256) fp32

    dim3 grid(LTOT / TILE, CH);               // (256, 128)
    dim3 block(TILE);                         // 256 threads = 8 wave32
    conv2d_local_gfx1250<<<grid, block, 0, stream>>>(x, w, o);
}